// GraphAttentionBlock_86337432584463
// MI455X (gfx1250) — compile-verified
//
#include <hip/hip_runtime.h>
#include <math.h>

typedef __attribute__((ext_vector_type(2))) float v2f;
typedef __attribute__((ext_vector_type(8))) float v8f;

#define HIDC 256
#define NHC  8
#define HDC  32
#define NSEQ 2048
#define NBAT 2

// ---------------------------------------------------------------------------
// LayerNorm: one wave (32 lanes) per 256-element row, 8 rows per 256-thr block
// ---------------------------------------------------------------------------
__global__ __launch_bounds__(256) void layernorm_k(
    const float* __restrict__ x, const float* __restrict__ g,
    const float* __restrict__ bta, float* __restrict__ out)
{
  const int lane = threadIdx.x & 31;
  const int wave = threadIdx.x >> 5;
  const int row  = blockIdx.x * 8 + wave;
  const float* xr = x + (size_t)row * HIDC;

  float v[8];
  float sum = 0.0f;
#pragma unroll
  for (int i = 0; i < 8; ++i) { v[i] = xr[lane + 32 * i]; sum += v[i]; }
#pragma unroll
  for (int m = 1; m < 32; m <<= 1) sum += __shfl_xor(sum, m, 32);
  const float mean = sum * (1.0f / HIDC);

  float vs = 0.0f;
#pragma unroll
  for (int i = 0; i < 8; ++i) { float d = v[i] - mean; vs += d * d; }
#pragma unroll
  for (int m = 1; m < 32; m <<= 1) vs += __shfl_xor(vs, m, 32);
  const float rstd = rsqrtf(vs * (1.0f / HIDC) + 1e-5f);

  float* orow = out + (size_t)row * HIDC;
#pragma unroll
  for (int i = 0; i < 8; ++i) {
    int c = lane + 32 * i;
    orow[c] = (v[i] - mean) * rstd * g[c] + bta[c];
  }
}

// ---------------------------------------------------------------------------
// f32 WMMA GEMM: out[M,N] = act(A[M,K] @ W[K,N] + bias) (+ resid)
// block = 128 thr (4 waves); wave tile = 32x64 (2 m-subtiles x 4 n-subtiles,
// 8 v8f accumulators) -> 1 global B load + 0.25 DS read per wmma.
// block tile = 32 rows x 256 cols.  A tile staged via float4 into LDS with
// stride 68 (16B-aligned rows, fragment reads hit banks 4*lr+k mod 64, halves
// offset by 2 mod 4 => conflict-free across all 32 lanes).
// ---------------------------------------------------------------------------
__global__ __launch_bounds__(128) void gemm_f32_wmma(
    const float* __restrict__ A, const float* __restrict__ W,
    const float* __restrict__ bias, const float* __restrict__ resid,
    float* __restrict__ out, int M, int N, int K, int act)
{
  __shared__ __align__(16) float As[32][68];
  const int tid  = threadIdx.x;
  const int lane = tid & 31;
  const int wave = tid >> 5;                 // 0..3 -> n subgroup
  const int m0   = blockIdx.x * 32;
  const int n0   = blockIdx.y * 256 + wave * 64;
  const int half = lane >> 4;
  const int lr   = lane & 15;                // frag row / C column
  const int off  = half * 2;                 // K offset for upper half-wave

  v8f c[2][4];
#pragma unroll
  for (int j = 0; j < 4; ++j) {
    const float bv = bias[n0 + 16 * j + lr];
#pragma unroll
    for (int i = 0; i < 2; ++i)
#pragma unroll
      for (int r = 0; r < 8; ++r) c[i][j][r] = bv;
  }

  for (int k0 = 0; k0 < K; k0 += 64) {
    // stage A tile 32x64 as float4 (512 vec4 loads / 128 threads = 4 each)
#pragma unroll
    for (int j = 0; j < 4; ++j) {
      int idx = tid + j * 128;
      int r = idx >> 4, c4 = (idx & 15) * 4;
      *(float4*)&As[r][c4] =
          *(const float4*)&A[(size_t)(m0 + r) * K + k0 + c4];
    }
    __syncthreads();
#pragma unroll 4
    for (int kk = 0; kk < 64; kk += 4) {
      v2f a[2], b[4];
#pragma unroll
      for (int i = 0; i < 2; ++i) {
        a[i].x = As[i * 16 + lr][kk + off];
        a[i].y = As[i * 16 + lr][kk + off + 1];
      }
      const float* wp = W + (size_t)(k0 + kk + off) * N + n0 + lr;
#pragma unroll
      for (int j = 0; j < 4; ++j) {
        b[j].x = wp[16 * j];
        b[j].y = wp[16 * j + N];
      }
#pragma unroll
      for (int i = 0; i < 2; ++i)
#pragma unroll
        for (int j = 0; j < 4; ++j)
          c[i][j] = __builtin_amdgcn_wmma_f32_16x16x4_f32(
              false, a[i], false, b[j], (short)0, c[i][j], false, false);
    }
    __syncthreads();
  }

#pragma unroll
  for (int i = 0; i < 2; ++i)
#pragma unroll
    for (int j = 0; j < 4; ++j)
#pragma unroll
      for (int r = 0; r < 8; ++r) {
        const int row = m0 + i * 16 + r + half * 8;
        const size_t oi = (size_t)row * N + n0 + 16 * j + lr;
        float x = c[i][j][r];
        if (act) x = 0.5f * x * (1.0f + erff(x * 0.70710678118654752f));
        if (resid) x += resid[oi];
        out[oi] = x;
      }
}

// ---------------------------------------------------------------------------
// Flash-style graph attention.
//   logits = QK^T/sqrt(32) + (2a-1);  p = exp(logit - m) * (0.25 + 0.75a)
//   O = sum(p*V) / sum(p)   (softmax denominator cancels under the renorm;
//   the 1e-6 clamp can't bind since the rescale factor >= 0.25)
// block = 4 waves x 16 q-rows; key tiles of 32 (two 16-col S subtiles per
// staging round).  K/V tiles staged as float4, stride 36 (conflict-free for
// the S-phase B-fragment reads, 16B-aligned rows).
// ---------------------------------------------------------------------------
__global__ __launch_bounds__(128) void graph_attn(
    const float* __restrict__ Q, const float* __restrict__ Kg,
    const float* __restrict__ V, const float* __restrict__ adj,
    float* __restrict__ outAtt)
{
  __shared__ __align__(16) float Ks[32][36];
  __shared__ __align__(16) float Vs[32][36];
  __shared__ float Ps[4][16][17];

  const int tid  = threadIdx.x;
  const int lane = tid & 31;
  const int wave = tid >> 5;
  const int h    = blockIdx.y;
  const int b    = blockIdx.z;
  const int q0   = blockIdx.x * 64 + wave * 16;
  const int half = lane >> 4;
  const int lr   = lane & 15;
  const int off  = half * 2;

  // Q strip (16 rows x 32 d) held as 8 A-fragments in registers
  v2f qa[8];
  const float* qrow = Q + (size_t)(b * NSEQ + q0 + lr) * HIDC + h * HDC;
#pragma unroll
  for (int s = 0; s < 8; ++s) {
    qa[s].x = qrow[4 * s + off];
    qa[s].y = qrow[4 * s + off + 1];
  }

  float mrow[8], lsum[8];
  v8f o1 = {}, o2 = {};
#pragma unroll
  for (int r = 0; r < 8; ++r) { mrow[r] = -1e30f; lsum[r] = 0.0f; }

  const float inv_sqrt_hd = 0.17677669529663687f;

  for (int kc = 0; kc < NSEQ; kc += 32) {
    // stage K,V 32x32 tiles as float4 (256 vec4 per array / 128 thr = 2 each)
#pragma unroll
    for (int j = 0; j < 2; ++j) {
      int idx = tid + j * 128;
      int r = idx >> 3, c4 = (idx & 7) * 4;
      size_t g = (size_t)(b * NSEQ + kc + r) * HIDC + h * HDC + c4;
      *(float4*)&Ks[r][c4] = *(const float4*)&Kg[g];
      *(float4*)&Vs[r][c4] = *(const float4*)&V[g];
    }
    __syncthreads();

#pragma unroll
    for (int kt = 0; kt < 2; ++kt) {
      const int kbase = kt * 16;

      // S = Q @ K^T  (contract over d = 32, 8 WMMA steps)
      v8f s = {};
#pragma unroll
      for (int st = 0; st < 8; ++st) {
        v2f bf;
        bf.x = Ks[kbase + lr][4 * st + off];
        bf.y = Ks[kbase + lr][4 * st + off + 1];
        s = __builtin_amdgcn_wmma_f32_16x16x4_f32(false, qa[st], false, bf,
                                                  (short)0, s, false, false);
      }

      // bias + online softmax update (row reductions across 16-lane halves)
#pragma unroll
      for (int r = 0; r < 8; ++r) {
        const int qg = q0 + r + half * 8;
        const float a = adj[(size_t)(b * NSEQ + qg) * NSEQ + kc + kbase + lr];
        const float logit = s[r] * inv_sqrt_hd + (2.0f * a - 1.0f);
        float tm = logit;
#pragma unroll
        for (int ms = 1; ms < 16; ms <<= 1) tm = fmaxf(tm, __shfl_xor(tm, ms, 32));
        const float nm = fmaxf(mrow[r], tm);
        const float sc = __expf(mrow[r] - nm);
        mrow[r] = nm;
        const float p = __expf(logit - nm) * (0.25f + 0.75f * a);
        float ps = p;
#pragma unroll
        for (int ms = 1; ms < 16; ms <<= 1) ps += __shfl_xor(ps, ms, 32);
        lsum[r] = lsum[r] * sc + ps;
        o1[r] *= sc;
        o2[r] *= sc;
        Ps[wave][r + half * 8][lr] = p;   // C-layout -> A-layout via LDS
      }
      // Ps is per-wave; same-wave LDS ops are in-order, no barrier needed.

      // O += P @ V  (contract over 16 key cols, two 16-wide d groups)
#pragma unroll
      for (int st = 0; st < 4; ++st) {
        v2f ap, bv1, bv2;
        ap.x  = Ps[wave][lr][4 * st + off];
        ap.y  = Ps[wave][lr][4 * st + off + 1];
        bv1.x = Vs[kbase + 4 * st + off][lr];
        bv1.y = Vs[kbase + 4 * st + off + 1][lr];
        bv2.x = Vs[kbase + 4 * st + off][16 + lr];
        bv2.y = Vs[kbase + 4 * st + off + 1][16 + lr];
        o1 = __builtin_amdgcn_wmma_f32_16x16x4_f32(false, ap, false, bv1,
                                                   (short)0, o1, false, false);
        o2 = __builtin_amdgcn_wmma_f32_16x16x4_f32(false, ap, false, bv2,
                                                   (short)0, o2, false, false);
      }
    }
    __syncthreads();
  }

#pragma unroll
  for (int r = 0; r < 8; ++r) {
    const int qg = q0 + r + half * 8;
    const float inv = 1.0f / lsum[r];     // lsum >= 0.25 always
    const size_t oi = (size_t)(b * NSEQ + qg) * HIDC + h * HDC;
    outAtt[oi + lr]      = o1[r] * inv;
    outAtt[oi + 16 + lr] = o2[r] * inv;
  }
}

// ---------------------------------------------------------------------------
extern "C" void kernel_launch(void* const* d_in, const int* in_sizes, int n_in,
                              void* d_out, int out_size, void* d_ws, size_t ws_size,
                              hipStream_t stream)
{
  const float* hidden = (const float*)d_in[0];
  const float* adj    = (const float*)d_in[1];
  // d_in[2] = positions (unused by the reference math)
  const float* Wq  = (const float*)d_in[3];
  const float* bq  = (const float*)d_in[4];
  const float* Wk  = (const float*)d_in[5];
  const float* bk  = (const float*)d_in[6];
  const float* Wv  = (const float*)d_in[7];
  const float* bv  = (const float*)d_in[8];
  const float* Wo  = (const float*)d_in[9];
  const float* bo  = (const float*)d_in[10];
  const float* g1  = (const float*)d_in[11];
  const float* b1  = (const float*)d_in[12];
  const float* g2  = (const float*)d_in[13];
  const float* b2  = (const float*)d_in[14];
  const float* Wf1 = (const float*)d_in[15];
  const float* bf1 = (const float*)d_in[16];
  const float* Wf2 = (const float*)d_in[17];
  const float* bf2 = (const float*)d_in[18];

  const int M = NBAT * NSEQ;               // 4096 rows
  float* ws = (float*)d_ws;
  float* normed   = ws; ws += (size_t)M * HIDC;
  float* Qb       = ws; ws += (size_t)M * HIDC;
  float* Kb       = ws; ws += (size_t)M * HIDC;
  float* Vb       = ws; ws += (size_t)M * HIDC;
  float* attended = ws; ws += (size_t)M * HIDC;
  float* hidden2  = ws; ws += (size_t)M * HIDC;
  float* ffin     = ws; ws += (size_t)M * HIDC;
  float* mid      = ws; ws += (size_t)M * 4 * HIDC;

  dim3 lnGrid(M / 8);
  layernorm_k<<<lnGrid, 256, 0, stream>>>(hidden, g1, b1, normed);

  dim3 gBlk(128);
  dim3 gGrid(M / 32, HIDC / 256);          // (128, 1)
  gemm_f32_wmma<<<gGrid, gBlk, 0, stream>>>(normed, Wq, bq, nullptr, Qb, M, HIDC, HIDC, 0);
  gemm_f32_wmma<<<gGrid, gBlk, 0, stream>>>(normed, Wk, bk, nullptr, Kb, M, HIDC, HIDC, 0);
  gemm_f32_wmma<<<gGrid, gBlk, 0, stream>>>(normed, Wv, bv, nullptr, Vb, M, HIDC, HIDC, 0);

  dim3 aGrid(NSEQ / 64, NHC, NBAT);
  graph_attn<<<aGrid, 128, 0, stream>>>(Qb, Kb, Vb, adj, attended);

  gemm_f32_wmma<<<gGrid, gBlk, 0, stream>>>(attended, Wo, bo, hidden, hidden2, M, HIDC, HIDC, 0);

  layernorm_k<<<lnGrid, 256, 0, stream>>>(hidden2, g2, b2, ffin);

  dim3 f1Grid(M / 32, (4 * HIDC) / 256);   // (128, 4)
  gemm_f32_wmma<<<f1Grid, gBlk, 0, stream>>>(ffin, Wf1, bf1, nullptr, mid, M, 4 * HIDC, HIDC, 1);

  gemm_f32_wmma<<<gGrid, gBlk, 0, stream>>>(mid, Wf2, bf2, hidden2, (float*)d_out, M, HIDC, 4 * HIDC, 0);
}